// Loss_19224273617388
// MI455X (gfx1250) — compile-verified
//
#include <hip/hip_runtime.h>

// ---------------------------------------------------------------------------
// Soft-DTW (soft weak-sequence-alignment) wavefront DP on MI455X (gfx1250).
//
// tc[i,j] = cost[i,j] + softmin_g(tc[i-1,j], tc[i,j-1], tc[i-1,j-1]), g=0.1
// Output: tc[R-1,C-1] (single fp32 scalar).
//
// Latency-bound: 8191 strictly serial anti-diagonals. Design:
//  * single 1024-thread workgroup (32 wave32) on one WGP; one barrier/diag
//  * thread t owns rows 4t..4t+3 -> left/up/diag for 3 of 4 rows stay in
//    registers; only the boundary row crosses threads via an LDS depth-3 ring
//  * cost anti-diagonals staged TWO ahead via global_load_async_to_lds_b32
//    into a triple buffer; exactly 4 async instructions per wave per diag
//    (column clamped -> uniform count), so s_wait_asynccnt 4 proves the
//    older diagonal landed while the newer one stays in flight
//  * softmin with exact min/median/max network: 2x v_exp_f32 + 1x v_log_f32
//    per cell (trans is the throughput bound at 1/4 rate)
// ---------------------------------------------------------------------------

#define R     4096
#define C     4096
#define NT    1024          // 32 waves of 32
#define RPT   4             // rows per thread; NT*RPT == R
#define DIAGS (R + C - 1)
#define BIGF  1e9f

#if __has_builtin(__builtin_amdgcn_exp2f)
#define EXP2F(x) __builtin_amdgcn_exp2f(x)
#else
#define EXP2F(x) exp2f(x)
#endif
#if __has_builtin(__builtin_amdgcn_logf)
#define LOG2F(x) __builtin_amdgcn_logf(x)   // v_log_f32 is log2
#else
#define LOG2F(x) log2f(x)
#endif

// CDNA5 async copy: global -> LDS, tracked by ASYNCcnt (no VGPR dest).
__device__ __forceinline__ void async_load_f32(const float* g, unsigned lds_byte) {
    asm volatile("global_load_async_to_lds_b32 %0, %1, off"
                 :: "v"(lds_byte), "v"(g)
                 : "memory");
}

// Allow `n` newest per-wave async instructions to remain outstanding.
#define WAIT_ASYNC(n) asm volatile("s_wait_asynccnt " #n ::: "memory")

// Stable softmin_g(a,b,c): the min term is exp2(0)==1, so only 2 exps needed.
// min/median/max via exact v_min/v_max network (no rounding error).
__device__ __forceinline__ float softmin3(float a, float b, float c) {
    const float kS = 14.426950408889634f;    // log2(e) / gamma, gamma = 0.1
    const float kG = 0.069314718055994531f;  // gamma * ln(2)
    float lo  = fminf(a, b);
    float hi  = fmaxf(a, b);
    float m   = fminf(lo, c);                 // min3
    float M   = fmaxf(hi, c);                 // max3
    float mid = fminf(hi, fmaxf(lo, c));      // exact median3
    float e1  = EXP2F((m - mid) * kS);        // in (0,1]
    float e2  = EXP2F((m - M) * kS);          // in (0,1]
    return m - kG * LOG2F(1.0f + e1 + e2);
}

__device__ __forceinline__ int clampj(int j) {
    return j < 0 ? 0 : (j > (C - 1) ? (C - 1) : j);
}

__global__ __launch_bounds__(NT) void dwsa_wavefront(const float* __restrict__ cost,
                                                     float* __restrict__ out) {
    __shared__ float cbuf[3][R];   // staged cost anti-diagonals (triple buffer)
    __shared__ float bnd[3][NT];   // boundary-row tc values, depth-3 ring

    const int t  = threadIdx.x;
    const int i0 = t * RPT;

    bnd[0][t] = BIGF;
    bnd[1][t] = BIGF;
    bnd[2][t] = BIGF;

    // prime the pipeline: stage diagonals 0 and 1 (4 async ops per wave each)
#pragma unroll
    for (int pd = 0; pd < 2; ++pd) {
#pragma unroll
        for (int k = 0; k < RPT; ++k) {
            const int i = i0 + k;
            const int j = clampj(pd - i);
            async_load_f32(cost + (size_t)i * C + (size_t)j,
                           (unsigned)(size_t)&cbuf[pd][i]);
        }
    }
    __syncthreads();

    float a1[RPT], a2[RPT];        // own tc on diagonals d-1, d-2
#pragma unroll
    for (int k = 0; k < RPT; ++k) { a1[k] = BIGF; a2[k] = BIGF; }

    // bnd ring: s0 = d%3 (write), s1 = (d+2)%3 (read d-1), s2 = (d+1)%3 (read d-2)
    int s0 = 0, s1 = 2, s2 = 1;
    // cbuf ring: pb = d%3 (read), pb2 = (d+2)%3 (stage target)
    int pb = 0, pb2 = 2;

    for (int d = 0; d < DIAGS; ++d) {
        // neighbor thread's boundary row (row i0-1) on diagonals d-1 / d-2
        float up_prev = (t > 0) ? bnd[s1][t - 1] : BIGF;
        float dg_prev = (t > 0) ? bnd[s2][t - 1] : BIGF;

        // 4 newest async ops (diag d+1) may stay in flight; diag d has landed
        WAIT_ASYNC(0x4);

        float cc[RPT];
#pragma unroll
        for (int k = 0; k < RPT; ++k) cc[k] = cbuf[pb][i0 + k];

        // stage diagonal d+2 (unconditional, clamped -> uniform 4 per wave)
        const int dnn = d + 2;
#pragma unroll
        for (int k = 0; k < RPT; ++k) {
            const int i = i0 + k;
            const int j = clampj(dnn - i);
            async_load_f32(cost + (size_t)i * C + (size_t)j,
                           (unsigned)(size_t)&cbuf[pb2][i]);
        }

        // compute this thread's 4 cells of diagonal d (independent -> ILP)
        float n[RPT];
#pragma unroll
        for (int k = 0; k < RPT; ++k) {
            const int i = i0 + k;
            const int j = d - i;
            float up   = (k == 0) ? up_prev : a1[k - 1];
            float dg   = (k == 0) ? dg_prev : a2[k - 1];
            float left = a1[k];
            if (i == 0) { up = BIGF;  dg = BIGF; }
            if (j == 0) { left = BIGF; dg = BIGF; }
            float v = cc[k] + softmin3(up, left, dg);
            if (i == 0 && j == 0) v = cc[k];       // tc[0,0] = cost[0,0]
            const bool valid = (j >= 0) && (j < C);
            n[k] = valid ? v : BIGF;
        }
#pragma unroll
        for (int k = 0; k < RPT; ++k) { a2[k] = a1[k]; a1[k] = n[k]; }

        // publish boundary row for thread t+1
        bnd[s0][t] = a1[RPT - 1];

        // rotate rings: x0' = x2, x1' = x0, x2' = x1 pattern
        {
            const int tmp = s0; s0 = s2; s2 = s1; s1 = tmp;
        }
        {
            const int tmp = pb;            // pb' = (d+1)%3, pb2' = (d+3)%3 = pb
            pb = (pb == 2) ? 0 : pb + 1;
            pb2 = tmp;
        }

        __syncthreads();   // exactly one barrier per diagonal
    }

    // after the last diagonal (d = R+C-2), thread NT-1 holds tc[R-1,C-1]
    if (t == NT - 1) out[0] = a1[RPT - 1];
}

extern "C" void kernel_launch(void* const* d_in, const int* in_sizes, int n_in,
                              void* d_out, int out_size, void* d_ws, size_t ws_size,
                              hipStream_t stream) {
    (void)in_sizes; (void)n_in; (void)d_ws; (void)ws_size; (void)out_size;
    const float* cost = (const float*)d_in[0];   // costV/costT are unused by the math
    float* out = (float*)d_out;
    dwsa_wavefront<<<1, NT, 0, stream>>>(cost, out);
}